// tinyLayerE_83975200571524
// MI455X (gfx1250) — compile-verified
//
#include <hip/hip_runtime.h>
#include <hip/hip_bf16.h>
#include <stdint.h>

// ---------------- sizes ----------------
#define BATCH   4096
#define DDIM    20000
#define BINS    10
#define KOUT    64          // k rows of samples
#define Y_ELEMS (BATCH * KOUT)          // 262144
#define S_ELEMS (KOUT * DDIM)           // 1280000

// GEMM tiling
#define MBLK    64           // rows of X per block (4 waves x 16)
#define KC      40           // k-chunk (per partition 2000 = 40*50)
#define KPART   10           // K-split partitions (deterministic partials)
#define KPER    (DDIM / KPART)   // 2000
#define NCHUNK  (KPER / KC)      // 50
#define LSTR    44           // padded LDS row stride (floats): 44r mod 64 distinct for r=0..15; 44*4B % 16B == 0
#define XOFF    0
#define SOFF    (MBLK * LSTR)
#define BUFSZ   ((MBLK + KOUT) * LSTR)   // 5632 floats = 22528 B per buffer

typedef float v2f __attribute__((ext_vector_type(2)));
typedef float v8f __attribute__((ext_vector_type(8)));
typedef int   v4i __attribute__((ext_vector_type(4)));

#define GAS __attribute__((address_space(1)))
#define LAS __attribute__((address_space(3)))

#if __has_builtin(__builtin_amdgcn_global_load_async_to_lds_b128)
#define HAVE_ASYNC_LDS 1
#else
#define HAVE_ASYNC_LDS 0
#endif

__device__ __forceinline__ void wait_async0() {
#if __has_builtin(__builtin_amdgcn_s_wait_asynccnt)
    __builtin_amdgcn_s_wait_asynccnt(0);
#else
    asm volatile("s_wait_asynccnt 0" ::: "memory");
#endif
}

// ---------------- small helpers ----------------
__device__ __forceinline__ unsigned hash_u32(unsigned x) {
    // counter hash keyed with 42 (stand-in for jax threefry; no runtime check)
    x ^= 42u * 0x9E3779B9u;
    x ^= x >> 16; x *= 0x7feb352dU;
    x ^= x >> 15; x *= 0x846ca68bU;
    x ^= x >> 16;
    return x;
}

__device__ __forceinline__ unsigned long long umax64(unsigned long long a, unsigned long long b) {
    return a > b ? a : b;
}

// ---------------- kernel 0: init ----------------
__global__ void init_kernel(float* __restrict__ discrete, float* __restrict__ newTempOut,
                            const float* __restrict__ temp,
                            unsigned* __restrict__ rowMask, unsigned* __restrict__ colMask,
                            unsigned long long* __restrict__ slot) {
    int idx = blockIdx.x * blockDim.x + threadIdx.x;
    for (int i = idx; i < S_ELEMS; i += gridDim.x * blockDim.x) {
        discrete[i] = 0.0f;
        if (i < DDIM) colMask[i] = 0u;
        if (i < KOUT) rowMask[i] = 0u;
        if (i == 0) {
            *slot = 0ull;
            newTempOut[0] = fmaxf(0.1f, temp[0] * 0.99999f);
        }
    }
}

// ---------------- kernel 1: logits+gumbel -> g (k,d), transposed ----------------
__global__ void k1_logits(const float* __restrict__ u, const float* __restrict__ W,
                          const float* __restrict__ temp, float* __restrict__ g) {
    __shared__ float wl[BINS * KOUT];   // 640 floats
    for (int i = threadIdx.x; i < BINS * KOUT; i += blockDim.x) wl[i] = W[i];
    __syncthreads();

    int d = blockIdx.x * blockDim.x + threadIdx.x;
    if (d >= DDIM) return;

    float newT = fmaxf(0.1f, temp[0] * 0.99999f);
    float ub[BINS];
#pragma unroll
    for (int b = 0; b < BINS; ++b) ub[b] = u[d * BINS + b];

    float s[KOUT];
    float mx = -1e30f;
#pragma unroll
    for (int k = 0; k < KOUT; ++k) {
        float acc = 0.0f;
#pragma unroll
        for (int b = 0; b < BINS; ++b) acc = fmaf(ub[b], wl[b * KOUT + k], acc);
        s[k] = acc;
        mx = fmaxf(mx, acc);
    }
    float sum = 0.0f;
#pragma unroll
    for (int k = 0; k < KOUT; ++k) { s[k] = __expf(s[k] - mx); sum += s[k]; }
    float inv = 1.0f / sum;

#pragma unroll
    for (int k = 0; k < KOUT; ++k) {
        float al = s[k] * inv;
        al = fminf(fmaxf(al, 1e-7f), 0.9999999f);
        float logit = __logf(10.0f * al);
        unsigned h = hash_u32((unsigned)(k * DDIM + d));
        float r = (float)(h >> 8) * (1.0f / 16777216.0f);      // [0,1)
        float uu = 1e-7f + (1.0f - 1e-7f) * r;                 // (eps,1)
        float gum = -__logf(-__logf(uu));
        g[k * DDIM + d] = (logit + gum) / newT;
    }
}

// ---------------- kernel 2: per-k max & sum(exp) over d ----------------
__global__ void k2_rowstats(const float* __restrict__ g, float* __restrict__ rowMax,
                            float* __restrict__ rowSum) {
    __shared__ float red[256];
    int k = blockIdx.x;
    const float* row = g + (size_t)k * DDIM;

    float mx = -1e30f;
    for (int d = threadIdx.x; d < DDIM; d += blockDim.x) mx = fmaxf(mx, row[d]);
    red[threadIdx.x] = mx;
    __syncthreads();
    for (int s = 128; s > 0; s >>= 1) {
        if (threadIdx.x < s) red[threadIdx.x] = fmaxf(red[threadIdx.x], red[threadIdx.x + s]);
        __syncthreads();
    }
    float smax = red[0];
    __syncthreads();

    float sum = 0.0f;
    for (int d = threadIdx.x; d < DDIM; d += blockDim.x) sum += __expf(row[d] - smax);
    red[threadIdx.x] = sum;
    __syncthreads();
    for (int s = 128; s > 0; s >>= 1) {
        if (threadIdx.x < s) red[threadIdx.x] += red[threadIdx.x + s];
        __syncthreads();
    }
    if (threadIdx.x == 0) { rowMax[k] = smax; rowSum[k] = red[0]; }
}

// ---------------- kernel 3: samples = exp(g-max)/sum, in place ----------------
__global__ void k3_samples(float* __restrict__ gs, const float* __restrict__ rowMax,
                           const float* __restrict__ rowSum) {
    int idx = blockIdx.x * blockDim.x + threadIdx.x;
    for (int i = idx; i < S_ELEMS; i += gridDim.x * blockDim.x) {
        int k = i / DDIM;
        gs[i] = __expf(gs[i] - rowMax[k]) / rowSum[k];
    }
}

// ---------------- kernel 4: WMMA GEMM, async double-buffered LDS ----------------
// grid: (BATCH/MBLK, KPART); 128 threads = 4 waves; wave w owns rows [w*16, w*16+16)
__global__ __launch_bounds__(128) void gemm_wmma(const float* __restrict__ X,
                                                 const float* __restrict__ S,
                                                 float* __restrict__ part) {
    __shared__ __align__(16) float lbuf[2][BUFSZ];   // 2 x 22528 B = 45 KB

    const int tid  = threadIdx.x;
    const int wave = tid >> 5;
    const int lane = tid & 31;
    const int lm   = lane & 15;
    const int lhi  = lane >> 4;

    const int mblk  = blockIdx.x * MBLK;
    const int kbase = blockIdx.y * KPER;

    v8f acc0 = {}, acc1 = {}, acc2 = {}, acc3 = {};

#if HAVE_ASYNC_LDS
    // async direct global->LDS staging of one chunk (X: 64x10 float4, S: 64x10 float4)
    auto stage_async = [&](int kc, float* buf) {
#pragma unroll
        for (int j = 0; j < 5; ++j) {
            int id = tid + j * 128;
            int r = id / 10, c = id % 10;
            const float* src = X + (size_t)(mblk + r) * DDIM + kc + c * 4;
            float* dst = buf + XOFF + r * LSTR + c * 4;
            __builtin_amdgcn_global_load_async_to_lds_b128(
                (GAS v4i*)(GAS void*)(void*)src,
                (LAS v4i*)(LAS void*)dst, 0, 0);
        }
#pragma unroll
        for (int j = 0; j < 5; ++j) {
            int id = tid + j * 128;
            int r = id / 10, c = id % 10;
            const float* src = S + (size_t)r * DDIM + kc + c * 4;
            float* dst = buf + SOFF + r * LSTR + c * 4;
            __builtin_amdgcn_global_load_async_to_lds_b128(
                (GAS v4i*)(GAS void*)(void*)src,
                (LAS v4i*)(LAS void*)dst, 0, 0);
        }
    };

    stage_async(kbase, lbuf[0]);
    for (int ch = 0; ch < NCHUNK; ++ch) {
        wait_async0();        // chunk ch landed in LDS (this wave's share)
        __syncthreads();      // all waves' shares visible; prev compute done -> other buf free
        if (ch + 1 < NCHUNK) stage_async(kbase + (ch + 1) * KC, lbuf[(ch + 1) & 1]);

        const float* buf = lbuf[ch & 1];
        const float* xa  = buf + XOFF + (wave * 16 + lm) * LSTR;
        const float* sb  = buf + SOFF + lm * LSTR;
#pragma unroll
        for (int ks = 0; ks < KC; ks += 4) {
            float2 av = *(const float2*)(xa + ks + lhi * 2);
            v2f a; a.x = av.x; a.y = av.y;
            const float* bb = sb + ks + lhi * 2;
            float2 b0v = *(const float2*)(bb);
            float2 b1v = *(const float2*)(bb + 16 * LSTR);
            float2 b2v = *(const float2*)(bb + 32 * LSTR);
            float2 b3v = *(const float2*)(bb + 48 * LSTR);
            v2f b0; b0.x = b0v.x; b0.y = b0v.y;
            v2f b1; b1.x = b1v.x; b1.y = b1v.y;
            v2f b2; b2.x = b2v.x; b2.y = b2v.y;
            v2f b3; b3.x = b3v.x; b3.y = b3v.y;
            acc0 = __builtin_amdgcn_wmma_f32_16x16x4_f32(false, a, false, b0, (short)0, acc0, false, false);
            acc1 = __builtin_amdgcn_wmma_f32_16x16x4_f32(false, a, false, b1, (short)0, acc1, false, false);
            acc2 = __builtin_amdgcn_wmma_f32_16x16x4_f32(false, a, false, b2, (short)0, acc2, false, false);
            acc3 = __builtin_amdgcn_wmma_f32_16x16x4_f32(false, a, false, b3, (short)0, acc3, false, false);
        }
    }
#else
    // fallback: synchronous staging (global->VGPR->LDS), alternating buffers
    for (int ch = 0; ch < NCHUNK; ++ch) {
        const int kc = kbase + ch * KC;
        float* buf = lbuf[ch & 1];
#pragma unroll
        for (int j = 0; j < 5; ++j) {
            int id = tid + j * 128;
            int r = id / 10, c = id % 10;
            float4 v = *(const float4*)(X + (size_t)(mblk + r) * DDIM + kc + c * 4);
            ((float4*)(buf + XOFF + r * LSTR))[c] = v;
        }
#pragma unroll
        for (int j = 0; j < 5; ++j) {
            int id = tid + j * 128;
            int r = id / 10, c = id % 10;
            float4 v = *(const float4*)(S + (size_t)r * DDIM + kc + c * 4);
            ((float4*)(buf + SOFF + r * LSTR))[c] = v;
        }
        __syncthreads();
        if (ch + 1 < NCHUNK)
            __builtin_prefetch(X + (size_t)(mblk + (tid & 63)) * DDIM + kc + KC, 0, 0);

        const float* xa = buf + XOFF + (wave * 16 + lm) * LSTR;
        const float* sb = buf + SOFF + lm * LSTR;
#pragma unroll
        for (int ks = 0; ks < KC; ks += 4) {
            float2 av = *(const float2*)(xa + ks + lhi * 2);
            v2f a; a.x = av.x; a.y = av.y;
            const float* bb = sb + ks + lhi * 2;
            float2 b0v = *(const float2*)(bb);
            float2 b1v = *(const float2*)(bb + 16 * LSTR);
            float2 b2v = *(const float2*)(bb + 32 * LSTR);
            float2 b3v = *(const float2*)(bb + 48 * LSTR);
            v2f b0; b0.x = b0v.x; b0.y = b0v.y;
            v2f b1; b1.x = b1v.x; b1.y = b1v.y;
            v2f b2; b2.x = b2v.x; b2.y = b2v.y;
            v2f b3; b3.x = b3v.x; b3.y = b3v.y;
            acc0 = __builtin_amdgcn_wmma_f32_16x16x4_f32(false, a, false, b0, (short)0, acc0, false, false);
            acc1 = __builtin_amdgcn_wmma_f32_16x16x4_f32(false, a, false, b1, (short)0, acc1, false, false);
            acc2 = __builtin_amdgcn_wmma_f32_16x16x4_f32(false, a, false, b2, (short)0, acc2, false, false);
            acc3 = __builtin_amdgcn_wmma_f32_16x16x4_f32(false, a, false, b3, (short)0, acc3, false, false);
        }
        __syncthreads();
    }
#endif

    // C/D layout: VGPR v -> M = v (lanes 0-15) / v+8 (lanes 16-31); N = lane&15 (+16*tile)
    float* out = part + (size_t)blockIdx.y * Y_ELEMS;
    const int m0 = mblk + wave * 16 + lhi * 8;
#pragma unroll
    for (int v = 0; v < 8; ++v) {
        out[(size_t)(m0 + v) * KOUT + lm +  0] = acc0[v];
        out[(size_t)(m0 + v) * KOUT + lm + 16] = acc1[v];
        out[(size_t)(m0 + v) * KOUT + lm + 32] = acc2[v];
        out[(size_t)(m0 + v) * KOUT + lm + 48] = acc3[v];
    }
}

// ---------------- kernel 5: deterministic reduction of K-partials ----------------
__global__ void reduce_partials(const float* __restrict__ part, float* __restrict__ Y) {
    int i = blockIdx.x * blockDim.x + threadIdx.x;
    if (i >= Y_ELEMS) return;
    float s = 0.0f;
#pragma unroll
    for (int p = 0; p < KPART; ++p) s += part[(size_t)p * Y_ELEMS + i];
    Y[i] = s;
}

// ---------------- greedy assignment: masked argmax + apply ----------------
__global__ void argmax_step(const float* __restrict__ samples,
                            const unsigned* __restrict__ rowMask,
                            const unsigned* __restrict__ colMask,
                            unsigned long long* __restrict__ slot) {
    __shared__ unsigned long long red[256];
    const int k = blockIdx.y;
    if (rowMask[k]) return;

    unsigned long long best = 0ull;
    const int base = blockIdx.x * blockDim.x * 4 + threadIdx.x;
#pragma unroll
    for (int j = 0; j < 4; ++j) {
        int d = base + j * blockDim.x;
        if (d < DDIM && !colMask[d]) {
            unsigned flat = (unsigned)(k * DDIM + d);
            unsigned bits = __float_as_uint(samples[flat]);   // samples > 0 -> order-preserving
            unsigned long long key = ((unsigned long long)bits << 32) | (0xFFFFFFFFu - flat);
            best = umax64(best, key);
        }
    }
    red[threadIdx.x] = best;
    __syncthreads();
    for (int s = 128; s > 0; s >>= 1) {
        if (threadIdx.x < s) red[threadIdx.x] = umax64(red[threadIdx.x], red[threadIdx.x + s]);
        __syncthreads();
    }
    if (threadIdx.x == 0 && red[0]) atomicMax(slot, red[0]);
}

__global__ void apply_step(unsigned long long* __restrict__ slot,
                           unsigned* __restrict__ rowMask, unsigned* __restrict__ colMask,
                           float* __restrict__ discrete) {
    unsigned long long key = *slot;
    if (key) {
        unsigned flat = 0xFFFFFFFFu - (unsigned)(key & 0xFFFFFFFFu);
        unsigned x = flat / DDIM, y = flat % DDIM;
        rowMask[x] = 1u;
        colMask[y] = 1u;
        discrete[(size_t)x * DDIM + y] = 1.0f;
    }
    *slot = 0ull;
}

// ---------------- launch ----------------
extern "C" void kernel_launch(void* const* d_in, const int* in_sizes, int n_in,
                              void* d_out, int out_size, void* d_ws, size_t ws_size,
                              hipStream_t stream) {
    const float* X    = (const float*)d_in[0];   // (4096, 20000)
    const float* u    = (const float*)d_in[1];   // (20000, 10)
    const float* W    = (const float*)d_in[2];   // (10, 64)
    const float* temp = (const float*)d_in[3];   // scalar

    float* Y        = (float*)d_out;                       // 262144
    float* discrete = (float*)d_out + Y_ELEMS;             // 1280000
    float* newTemp  = (float*)d_out + Y_ELEMS + S_ELEMS;   // 1

    // workspace layout (floats); total ~3.92M floats (~15.7 MB)
    float* ws      = (float*)d_ws;
    float* gs      = ws;                         // g -> samples in place (1,280,000)
    float* rowMax  = ws + S_ELEMS;               // 64
    float* rowSum  = ws + S_ELEMS + 64;          // 64
    float* part    = ws + S_ELEMS + 128;         // 10 * 262144
    float* tail    = part + KPART * Y_ELEMS;     // byte offset % 8 == 0
    unsigned long long* slot = (unsigned long long*)tail;
    unsigned* rowMask = (unsigned*)(tail + 2);
    unsigned* colMask = (unsigned*)(tail + 2 + 64);

    // 0) init outputs / masks / slot / new_temp
    init_kernel<<<1250, 256, 0, stream>>>(discrete, newTemp, temp, rowMask, colMask, slot);

    // 1) g = (log(10*clip(softmax_k(u@W))) + gumbel) / new_temp, transposed (k,d)
    k1_logits<<<(DDIM + 255) / 256, 256, 0, stream>>>(u, W, temp, gs);

    // 2) per-k row stats over d
    k2_rowstats<<<KOUT, 256, 0, stream>>>(gs, rowMax, rowSum);

    // 3) samples in place
    k3_samples<<<1250, 256, 0, stream>>>(gs, rowMax, rowSum);

    // 4) Y partials = X @ samples^T via V_WMMA_F32_16X16X4_F32 (async double-buffered LDS)
    gemm_wmma<<<dim3(BATCH / MBLK, KPART), 128, 0, stream>>>(X, gs, part);

    // 5) deterministic partial reduction
    reduce_partials<<<(Y_ELEMS + 255) / 256, 256, 0, stream>>>(part, Y);

    // 6) greedy bipartite argmax-and-mask (64 sequential steps)
    for (int it = 0; it < KOUT; ++it) {
        argmax_step<<<dim3(20, KOUT), 256, 0, stream>>>(gs, rowMask, colMask, slot);
        apply_step<<<1, 1, 0, stream>>>(slot, rowMask, colMask, discrete);
    }
}